// MOLAttention_4045859193223
// MI455X (gfx1250) — compile-verified
//
#include <hip/hip_runtime.h>
#include <hip/hip_bf16.h>
#include <math.h>

#define B_SZ 64
#define T_ENC 1024
#define D_ENC 512
#define Q_DIM 1024
#define H_DIM 256
#define M_MIX 5
#define EPSF 1e-5f

typedef __attribute__((ext_vector_type(2))) float v2f;
typedef __attribute__((ext_vector_type(8))) float v8f;

// ---------------------------------------------------------------------------
// Kernel 1: h = relu(att_rnn_h @ W1 + b1)   (64x1024)@(1024x256) -> (64,256)
// One wave (32 lanes) per 16x16 output tile; K-loop of 256 steps of K=4 using
// v_wmma_f32_16x16x4_f32 (fp32 WMMA, matches fp32 inputs exactly).
// Tiles: 4 (M) x 16 (N) = 64 waves = 64 blocks of 32 threads.
// ---------------------------------------------------------------------------
__global__ __launch_bounds__(32)
void mol_mlp1_wmma(const float* __restrict__ A,   // (64,1024)
                   const float* __restrict__ W1,  // (1024,256)
                   const float* __restrict__ b1,  // (256)
                   float* __restrict__ h)         // (64,256)
{
    const int tile  = blockIdx.x;     // 0..63
    const int tileM = tile >> 4;      // 0..3
    const int tileN = tile & 15;      // 0..15
    const int lane  = threadIdx.x;    // 0..31 (wave32)
    const int half  = lane >> 4;      // 0: K=0,1   1: K=2,3
    const int l15   = lane & 15;

    const int mA   = tileM * 16 + l15;   // A row for this lane
    const int nB   = tileN * 16 + l15;   // B col for this lane
    const int koff = half * 2;

    const float* arow = A + (size_t)mA * Q_DIM + koff;  // A[mA][koff + k0 + {0,1}]

    v8f acc = {};
    #pragma unroll 4
    for (int k0 = 0; k0 < Q_DIM; k0 += 4) {
        v2f av, bv;
        av.x = arow[k0];
        av.y = arow[k0 + 1];
        bv.x = W1[(size_t)(k0 + koff    ) * H_DIM + nB];
        bv.y = W1[(size_t)(k0 + koff + 1) * H_DIM + nB];
        // D = A(16x4) x B(4x16) + C ; emits v_wmma_f32_16x16x4_f32
        acc = __builtin_amdgcn_wmma_f32_16x16x4_f32(
            /*neg_a=*/false, av, /*neg_b=*/false, bv,
            /*c_mod=*/(short)0, acc, /*reuse_a=*/false, /*reuse_b=*/false);
    }

    // C/D layout: VGPR r holds row (r + 8*half), col l15
    const float bias = b1[tileN * 16 + l15];
    #pragma unroll
    for (int r = 0; r < 8; ++r) {
        const int m = tileM * 16 + r + half * 8;
        float v = acc[r] + bias;
        h[(size_t)m * H_DIM + tileN * 16 + l15] = v > 0.0f ? v : 0.0f;
    }
}

// ---------------------------------------------------------------------------
// Kernel 2: per batch row b — params = h@W2 + b2, MOL param transforms,
// then alpha_t[b][0..1023].  One block (256 threads) per b.
// ---------------------------------------------------------------------------
__device__ __forceinline__ float softplusf(float x) {
    // numerically stable: max(x,0) + log1p(exp(-|x|))
    return fmaxf(x, 0.0f) + log1pf(__expf(-fabsf(x)));
}
__device__ __forceinline__ float sigmoidf(float z) {
    return 1.0f / (1.0f + __expf(-z));
}

__global__ __launch_bounds__(256)
void mol_params_alpha(const float* __restrict__ h,        // (64,256)
                      const float* __restrict__ W2,       // (256,15)
                      const float* __restrict__ b2,       // (15)
                      const float* __restrict__ mu_prev,  // (64,5)
                      const unsigned char* __restrict__ mask, // (64,1024) bool
                      float* __restrict__ alpha)          // (64,1024)
{
    const int b   = blockIdx.x;
    const int tid = threadIdx.x;

    __shared__ float hs[H_DIM];
    __shared__ float ps[16];

    hs[tid] = h[(size_t)b * H_DIM + tid];
    __syncthreads();

    if (tid < 3 * M_MIX) {
        float s = b2[tid];
        #pragma unroll 8
        for (int k = 0; k < H_DIM; ++k)
            s = fmaf(hs[k], W2[k * (3 * M_MIX) + tid], s);
        ps[tid] = s;
    }
    __syncthreads();

    // Each thread derives its private copy of (w, sigma, mu) — cheap.
    float w[M_MIX], sig[M_MIX], mu[M_MIX];
    float wmax = ps[0];
    #pragma unroll
    for (int i = 1; i < M_MIX; ++i) wmax = fmaxf(wmax, ps[i]);
    float wsum = 0.0f;
    #pragma unroll
    for (int i = 0; i < M_MIX; ++i) { w[i] = __expf(ps[i] - wmax); wsum += w[i]; }
    const float winv = 1.0f / wsum;
    #pragma unroll
    for (int i = 0; i < M_MIX; ++i) {
        w[i]   = w[i] * winv + EPSF;
        sig[i] = softplusf(ps[M_MIX + i]) + EPSF;
        mu[i]  = mu_prev[b * M_MIX + i] + softplusf(ps[2 * M_MIX + i]);
    }

    for (int t = tid; t < T_ENC; t += 256) {
        float F0 = 0.0f, F1 = 0.0f;
        const float j0 = (float)t + 0.5f;
        const float j1 = (float)t + 1.5f;
        #pragma unroll
        for (int i = 0; i < M_MIX; ++i) {
            const float is = 1.0f / sig[i];
            F0 += w[i] * (1.0f / (1.0f + sigmoidf((mu[i] - j0) * is)));
            F1 += w[i] * (1.0f / (1.0f + sigmoidf((mu[i] - j1) * is)));
        }
        float a = F1 - F0;
        if (a == 0.0f) a = EPSF;
        if (mask[(size_t)b * T_ENC + t]) a = 0.0f;
        alpha[(size_t)b * T_ENC + t] = a;
    }
}

// ---------------------------------------------------------------------------
// Kernel 3a: zero the context region (atomics accumulate into it).
// ---------------------------------------------------------------------------
__global__ __launch_bounds__(256)
void zero_ctx(float* __restrict__ ctx, int n)
{
    const int i = blockIdx.x * 256 + threadIdx.x;
    if (i < n) ctx[i] = 0.0f;
}

// ---------------------------------------------------------------------------
// Kernel 3b: context[b][d] = sum_t alpha[b][t] * memory[b][t][d]
// Bandwidth-bound: streams 134 MB once. Split-K over T (16 chunks of 64) for
// parallelism (64*16 = 1024 blocks), float4 loads (global_load_b128),
// global_prefetch_b8 hints, fp32 global atomics to combine partials.
// Block = 128 threads, each thread owns 4 consecutive d (covers all 512).
// ---------------------------------------------------------------------------
__global__ __launch_bounds__(128)
void mol_context(const float* __restrict__ alpha,   // (64,1024)
                 const float* __restrict__ memory,  // (64,1024,512)
                 float* __restrict__ ctx)           // (64,512)
{
    const int b   = blockIdx.x;          // 0..63
    const int tch = blockIdx.y;          // 0..15   (T chunk)
    const int d4  = threadIdx.x;         // 0..127  -> d = d4*4

    const float4* mem = (const float4*)(memory + (size_t)b * T_ENC * D_ENC);
    const float*  al  = alpha + (size_t)b * T_ENC;

    float4 acc = make_float4(0.0f, 0.0f, 0.0f, 0.0f);

    const int t0 = tch * (T_ENC / 16);
    const int t1 = t0 + (T_ENC / 16);
    for (int t = t0; t < t1; ++t) {
        // prefetch a few rows ahead (global_prefetch_b8)
        if (t + 8 < t1)
            __builtin_prefetch(&mem[(size_t)(t + 8) * (D_ENC / 4) + d4], 0, 0);
        const float a = al[t];                       // uniform -> scalar load
        const float4 m = mem[(size_t)t * (D_ENC / 4) + d4];
        acc.x = fmaf(a, m.x, acc.x);
        acc.y = fmaf(a, m.y, acc.y);
        acc.z = fmaf(a, m.z, acc.z);
        acc.w = fmaf(a, m.w, acc.w);
    }

    float* c = ctx + (size_t)b * D_ENC + d4 * 4;
    atomicAdd(c + 0, acc.x);
    atomicAdd(c + 1, acc.y);
    atomicAdd(c + 2, acc.z);
    atomicAdd(c + 3, acc.w);
}

// ---------------------------------------------------------------------------
// Launch. Input order: att_rnn_h, memory, mask, mu_prev, W1, b1, W2, b2.
// d_out = [context (64*512) | alpha (64*1024)] fp32.
// ---------------------------------------------------------------------------
extern "C" void kernel_launch(void* const* d_in, const int* in_sizes, int n_in,
                              void* d_out, int out_size, void* d_ws, size_t ws_size,
                              hipStream_t stream) {
    const float*         att_rnn_h = (const float*)d_in[0];
    const float*         memory    = (const float*)d_in[1];
    const unsigned char* mask      = (const unsigned char*)d_in[2];
    const float*         mu_prev   = (const float*)d_in[3];
    const float*         W1        = (const float*)d_in[4];
    const float*         b1        = (const float*)d_in[5];
    const float*         W2        = (const float*)d_in[6];
    const float*         b2        = (const float*)d_in[7];

    float* ctx_out   = (float*)d_out;                    // (64,512)
    float* alpha_out = (float*)d_out + B_SZ * D_ENC;     // (64,1024)
    float* h_ws      = (float*)d_ws;                     // (64,256) scratch

    // 1) fp32 WMMA GEMM: h = relu(A @ W1 + b1). 64 tiles, one wave each.
    mol_mlp1_wmma<<<64, 32, 0, stream>>>(att_rnn_h, W1, b1, h_ws);

    // 2) MOL params + alpha per batch row.
    mol_params_alpha<<<B_SZ, 256, 0, stream>>>(h_ws, W2, b2, mu_prev, mask, alpha_out);

    // 3) context reduction (zero first; split-K atomics accumulate).
    zero_ctx<<<(B_SZ * D_ENC + 255) / 256, 256, 0, stream>>>(ctx_out, B_SZ * D_ENC);
    mol_context<<<dim3(B_SZ, 16), 128, 0, stream>>>(alpha_out, memory, ctx_out);
}